// DrugGPS_26087631356274
// MI455X (gfx1250) — compile-verified
//
#include <hip/hip_runtime.h>
#include <hip/hip_bf16.h>

#define B_      4096
#define H_      256
#define P_      128
#define V_      2000
#define NATOMS_ 32768
#define NRES_   262144
#define KNN_    4

typedef __attribute__((ext_vector_type(2))) float v2f;
typedef __attribute__((ext_vector_type(8))) float v8f;

__device__ __forceinline__ int lower_bound_i(const int* __restrict__ a, int n, int key) {
    int lo = 0, hi = n;
    while (lo < hi) { int mid = (lo + hi) >> 1; if (a[mid] < key) lo = mid + 1; else hi = mid; }
    return lo;
}

// ---------------------------------------------------------------------------
// K1: segment sums over atoms (sorted batch): node_hiddens[B,H], center_pos[B,3]
// ---------------------------------------------------------------------------
__global__ void seg_atoms_kernel(const float* __restrict__ h_ctx,
                                 const float* __restrict__ pos,
                                 const int*   __restrict__ batch,
                                 float* __restrict__ node_hiddens,
                                 float* __restrict__ center_pos) {
    int b = blockIdx.x, t = threadIdx.x;
    int lo = lower_bound_i(batch, NATOMS_, b);
    int hi = lower_bound_i(batch, NATOMS_, b + 1);
    float acc = 0.0f;
    for (int r = lo; r < hi; ++r) acc += h_ctx[(size_t)r * H_ + t];
    node_hiddens[(size_t)b * H_ + t] = acc;
    if (t < 3) {
        float ps = 0.0f;
        for (int r = lo; r < hi; ++r) ps += pos[(size_t)r * 3 + t];
        float cnt = (float)(hi - lo);
        center_pos[(size_t)b * 3 + t] = ps / fmaxf(cnt, 1.0f);
    }
}

// ---------------------------------------------------------------------------
// K2a: per-prototype stats: pnorm[p] = max(||proto_p||,1e-12); inv_sp[p]=1/(rowsum+1)
// ---------------------------------------------------------------------------
__global__ void proto_stats_kernel(const float* __restrict__ prototypes,
                                   const float* __restrict__ pcount,
                                   float* __restrict__ pnorm,
                                   float* __restrict__ inv_sp) {
    __shared__ float red[256];
    int p = blockIdx.x, t = threadIdx.x;
    float v = prototypes[(size_t)p * H_ + t];
    red[t] = v * v;
    __syncthreads();
    for (int s = 128; s > 0; s >>= 1) { if (t < s) red[t] += red[t + s]; __syncthreads(); }
    if (t == 0) pnorm[p] = fmaxf(sqrtf(red[0]), 1e-12f);
    __syncthreads();
    float s2 = 0.0f;
    for (int vv = t; vv < V_; vv += 256) s2 += pcount[(size_t)p * V_ + vv];
    red[t] = s2;
    __syncthreads();
    for (int s = 128; s > 0; s >>= 1) { if (t < s) red[t] += red[t + s]; __syncthreads(); }
    if (t == 0) inv_sp[p] = 1.0f / (red[0] + 1.0f);
}

// ---------------------------------------------------------------------------
// K2b: per-vocab coefficient: coeff[v] = log((1+P)/(1+sum_motif[v])) + 1
// ---------------------------------------------------------------------------
__global__ void vocab_coeff_kernel(const float* __restrict__ pcount,
                                   float* __restrict__ coeff) {
    int v = blockIdx.x * blockDim.x + threadIdx.x;
    if (v >= V_) return;
    float s = 0.0f;
    for (int p = 0; p < P_; ++p) s += (pcount[(size_t)p * V_ + v] > 0.0f) ? 1.0f : 0.0f;
    coeff[v] = logf((1.0f + (float)P_) / (1.0f + s)) + 1.0f;
}

// ---------------------------------------------------------------------------
// K3: masked segment sum over residues (sorted batch) + row L2 norm.
// Dominant 256MB HBM stream; prefetch 8 rows ahead (global_prefetch_b8,
// speculative -> safely dropped past buffer end).
// ---------------------------------------------------------------------------
__global__ void seg_residue_kernel(const float* __restrict__ h_res,
                                   const float* __restrict__ rpos,
                                   const int*   __restrict__ batch,
                                   const float* __restrict__ center_pos,
                                   float* __restrict__ residue_emb,
                                   float* __restrict__ rnorm) {
    __shared__ float red[256];
    int b = blockIdx.x, t = threadIdx.x;
    int lo = lower_bound_i(batch, NRES_, b);
    int hi = lower_bound_i(batch, NRES_, b + 1);
    float cx = center_pos[(size_t)b * 3 + 0];
    float cy = center_pos[(size_t)b * 3 + 1];
    float cz = center_pos[(size_t)b * 3 + 2];
    float acc = 0.0f;
    for (int r = lo; r < hi; ++r) {
        __builtin_prefetch(h_res + (size_t)(r + 8) * H_ + t, 0, 1);
        // CA = residue_pos[r, 1, :]
        float dx = rpos[(size_t)r * 9 + 3] - cx;
        float dy = rpos[(size_t)r * 9 + 4] - cy;
        float dz = rpos[(size_t)r * 9 + 5] - cz;
        float d2 = dx * dx + dy * dy + dz * dz;
        if (d2 < 36.0f)  // dist < 6.0
            acc += h_res[(size_t)r * H_ + t];
    }
    residue_emb[(size_t)b * H_ + t] = acc;
    red[t] = acc * acc;
    __syncthreads();
    for (int s = 128; s > 0; s >>= 1) { if (t < s) red[t] += red[t + s]; __syncthreads(); }
    if (t == 0) rnorm[b] = fmaxf(sqrtf(red[0]), 1e-12f);
}

// ---------------------------------------------------------------------------
// K4: cosine vs prototypes + top-4 (ties -> lower index, matching lax.top_k)
// ---------------------------------------------------------------------------
__global__ void cos_topk_kernel(const float* __restrict__ residue_emb,
                                const float* __restrict__ prototypes,
                                const float* __restrict__ rnorm,
                                const float* __restrict__ pnorm,
                                int* __restrict__ knn) {
    __shared__ float re[H_];
    __shared__ float cosv[P_];
    int b = blockIdx.x, t = threadIdx.x;  // blockDim = 128
    re[t]       = residue_emb[(size_t)b * H_ + t];
    re[t + 128] = residue_emb[(size_t)b * H_ + t + 128];
    __syncthreads();
    const float* pr = prototypes + (size_t)t * H_;
    float dot = 0.0f;
    for (int j = 0; j < H_; ++j) dot += re[j] * pr[j];
    cosv[t] = dot / (rnorm[b] * pnorm[t]);
    __syncthreads();
    if (t == 0) {
        for (int k = 0; k < KNN_; ++k) {
            float best = -3.0e38f; int bi = 0;
            for (int p = 0; p < P_; ++p) { float c = cosv[p]; if (c > best) { best = c; bi = p; } }
            knn[b * KNN_ + k] = bi;
            cosv[bi] = -3.0e38f;
        }
    }
}

// ---------------------------------------------------------------------------
// K5: Yproto[p] = sum_{i: p in knn(i)} residue_emb[i] + block[p,:] @ E[:V]
// ---------------------------------------------------------------------------
__global__ void yproto_kernel(const float* __restrict__ residue_emb,
                              const int*   __restrict__ knn,
                              const float* __restrict__ pcount,
                              const float* __restrict__ coeff,
                              const float* __restrict__ inv_sp,
                              const float* __restrict__ embw,
                              float* __restrict__ Yproto) {
    int p = blockIdx.x, t = threadIdx.x;  // 256 threads
    float acc = 0.0f;
    for (int i = 0; i < B_; ++i) {
        const int* k4 = knn + i * KNN_;
        bool m = (k4[0] == p) | (k4[1] == p) | (k4[2] == p) | (k4[3] == p);
        if (m) acc += residue_emb[(size_t)i * H_ + t];
    }
    float isp = inv_sp[p];
    for (int v = 0; v < V_; ++v) {
        float bval = pcount[(size_t)p * V_ + v] * isp * coeff[v];
        acc += bval * embw[(size_t)v * H_ + t];
    }
    Yproto[(size_t)p * H_ + t] = acc;
}

// ---------------------------------------------------------------------------
// K8: pred_vecs[b] = [0.8*node_hiddens | 0.7*E[wid] | 1.5*sum_k hp2[knn_k]]
// ---------------------------------------------------------------------------
__global__ void predvec_kernel(const float* __restrict__ node_hiddens,
                               const float* __restrict__ embw,
                               const int*   __restrict__ wid,
                               const float* __restrict__ hp2,
                               const int*   __restrict__ knn,
                               float* __restrict__ pv) {
    int b = blockIdx.x, t = threadIdx.x;
    pv[(size_t)b * (3 * H_) + t]          = 0.8f * node_hiddens[(size_t)b * H_ + t];
    pv[(size_t)b * (3 * H_) + H_ + t]     = 0.7f * embw[(size_t)wid[b] * H_ + t];
    float s = 0.0f;
    for (int k = 0; k < KNN_; ++k) s += hp2[(size_t)knn[b * KNN_ + k] * H_ + t];
    pv[(size_t)b * (3 * H_) + 2 * H_ + t] = 1.5f * s;
}

// ---------------------------------------------------------------------------
// WMMA f32 GEMM, B row-major [K,N], compile-time K/N for strength reduction.
// One wave -> 16x64 output (4 accumulators); each A fetch feeds 4 WMMAs.
// A layout (16x4): lane(l<16): M=l, K=k0..k0+1 ; lane>=16: M=l-16, K=k0+2..k0+3
// C/D layout: vgpr j -> M = j + 8*(lane>=16), N = lane&15
// ---------------------------------------------------------------------------
template<int K, int N, bool RELU>
__global__ void gemm_wmma_kn_t(const float* __restrict__ X, const float* __restrict__ W,
                               const float* __restrict__ bias, float* __restrict__ Out,
                               int M) {
    constexpr int GN = N / 64;  // groups of 4 N-tiles
    int gwave = (blockIdx.x * blockDim.x + threadIdx.x) >> 5;
    int lane  = threadIdx.x & 31;
    int tm = gwave / GN;
    int tg = gwave - tm * GN;
    if (tm * 16 >= M) return;  // uniform per wave
    int half = lane >> 4, l15 = lane & 15;
    const float* xrow = X + (size_t)(tm * 16 + l15) * K;
    int col0 = tg * 64 + l15;
    const float* wcol = W + col0;
    v8f c0 = {}, c1 = {}, c2 = {}, c3 = {};
    for (int k0 = 0; k0 < K; k0 += 4) {
        int kk = k0 + half * 2;
        v2f a = *(const v2f*)(xrow + kk);
        const float* wp = wcol + (size_t)kk * N;
        v2f b0, b1, b2, b3;
        b0.x = wp[0];  b0.y = wp[N];
        b1.x = wp[16]; b1.y = wp[N + 16];
        b2.x = wp[32]; b2.y = wp[N + 32];
        b3.x = wp[48]; b3.y = wp[N + 48];
        c0 = __builtin_amdgcn_wmma_f32_16x16x4_f32(false, a, false, b0, (short)0, c0, false, false);
        c1 = __builtin_amdgcn_wmma_f32_16x16x4_f32(false, a, false, b1, (short)0, c1, false, false);
        c2 = __builtin_amdgcn_wmma_f32_16x16x4_f32(false, a, false, b2, (short)0, c2, false, false);
        c3 = __builtin_amdgcn_wmma_f32_16x16x4_f32(false, a, false, b3, (short)0, c3, false, false);
    }
    float bb0 = bias ? bias[col0]      : 0.0f;
    float bb1 = bias ? bias[col0 + 16] : 0.0f;
    float bb2 = bias ? bias[col0 + 32] : 0.0f;
    float bb3 = bias ? bias[col0 + 48] : 0.0f;
#pragma unroll
    for (int j = 0; j < 8; ++j) {
        float* orow = Out + (size_t)(tm * 16 + half * 8 + j) * N + col0;
        float v0 = c0[j] + bb0, v1 = c1[j] + bb1, v2 = c2[j] + bb2, v3 = c3[j] + bb3;
        if (RELU) {
            v0 = fmaxf(v0, 0.0f); v1 = fmaxf(v1, 0.0f);
            v2 = fmaxf(v2, 0.0f); v3 = fmaxf(v3, 0.0f);
        }
        orow[0] = v0; orow[16] = v1; orow[32] = v2; orow[48] = v3;
    }
}

// ---------------------------------------------------------------------------
// WMMA f32 GEMM, B transposed [Nreal,K] (contiguous b64 loads per B row).
// One wave -> 16x64 output; Nreal not a multiple of 64: loads clamped, stores
// guarded per subtile (WMMA itself always runs with full EXEC).
// ---------------------------------------------------------------------------
template<int K>
__global__ void gemm_wmma_nt_t(const float* __restrict__ X, const float* __restrict__ Wt,
                               float* __restrict__ Out, int M, int Nreal, int groupsN) {
    int gwave = (blockIdx.x * blockDim.x + threadIdx.x) >> 5;
    int lane  = threadIdx.x & 31;
    int tm = gwave / groupsN;
    int tg = gwave - tm * groupsN;
    if (tm * 16 >= M) return;
    int half = lane >> 4, l15 = lane & 15;
    const float* xrow = X + (size_t)(tm * 16 + l15) * K;
    int n0 = tg * 64 + l15;
    int n1 = n0 + 16, n2 = n0 + 32, n3 = n0 + 48;
    const float* w0 = Wt + (size_t)(n0 < Nreal ? n0 : Nreal - 1) * K;
    const float* w1 = Wt + (size_t)(n1 < Nreal ? n1 : Nreal - 1) * K;
    const float* w2 = Wt + (size_t)(n2 < Nreal ? n2 : Nreal - 1) * K;
    const float* w3 = Wt + (size_t)(n3 < Nreal ? n3 : Nreal - 1) * K;
    v8f c0 = {}, c1 = {}, c2 = {}, c3 = {};
    for (int k0 = 0; k0 < K; k0 += 4) {
        int kk = k0 + half * 2;
        v2f a  = *(const v2f*)(xrow + kk);
        v2f b0 = *(const v2f*)(w0 + kk);
        v2f b1 = *(const v2f*)(w1 + kk);
        v2f b2 = *(const v2f*)(w2 + kk);
        v2f b3 = *(const v2f*)(w3 + kk);
        c0 = __builtin_amdgcn_wmma_f32_16x16x4_f32(false, a, false, b0, (short)0, c0, false, false);
        c1 = __builtin_amdgcn_wmma_f32_16x16x4_f32(false, a, false, b1, (short)0, c1, false, false);
        c2 = __builtin_amdgcn_wmma_f32_16x16x4_f32(false, a, false, b2, (short)0, c2, false, false);
        c3 = __builtin_amdgcn_wmma_f32_16x16x4_f32(false, a, false, b3, (short)0, c3, false, false);
    }
#pragma unroll
    for (int j = 0; j < 8; ++j) {
        float* orow = Out + (size_t)(tm * 16 + half * 8 + j) * Nreal;
        if (n0 < Nreal) orow[n0] = c0[j];
        if (n1 < Nreal) orow[n1] = c1[j];
        if (n2 < Nreal) orow[n2] = c2[j];
        if (n3 < Nreal) orow[n3] = c3[j];
    }
}

// ---------------------------------------------------------------------------
extern "C" void kernel_launch(void* const* d_in, const int* in_sizes, int n_in,
                              void* d_out, int out_size, void* d_ws, size_t ws_size,
                              hipStream_t stream) {
    const float* h_ctx_focal   = (const float*)d_in[0];
    const float* pos_ctx_focal = (const float*)d_in[1];
    const float* h_residue     = (const float*)d_in[2];
    const float* residue_pos   = (const float*)d_in[3];
    const float* embw          = (const float*)d_in[4];   // (V+1, H)
    const float* prototypes    = (const float*)d_in[5];   // (P, H)
    const float* pcount        = (const float*)d_in[6];   // (P, V)
    const float* motif_w       = (const float*)d_in[7];   // (3H, H)
    const float* motif_b       = (const float*)d_in[8];   // (H,)
    const float* gnn_w1        = (const float*)d_in[9];   // (H, H)
    const float* gnn_w2        = (const float*)d_in[10];  // (H, H)
    const int*   current_wid   = (const int*)d_in[11];    // (B,)
    const int*   atoms_batch   = (const int*)d_in[12];    // (N_ATOMS,) sorted
    const int*   aa_batch      = (const int*)d_in[13];    // (N_RES,) sorted
    float* out = (float*)d_out;                           // (B, V+1)

    // workspace layout (floats)
    float* ws = (float*)d_ws;
    size_t o = 0;
    float* node_hiddens = ws + o; o += (size_t)B_ * H_;
    float* center_pos   = ws + o; o += (size_t)B_ * 3;
    float* rnorm        = ws + o; o += B_;
    float* pnorm        = ws + o; o += P_;
    float* inv_sp       = ws + o; o += P_;
    float* coeff        = ws + o; o += V_;
    float* residue_emb  = ws + o; o += (size_t)B_ * H_;
    float* Yproto       = ws + o; o += (size_t)P_ * H_;
    float* h1p          = ws + o; o += (size_t)P_ * H_;
    float* hp2          = ws + o; o += (size_t)P_ * H_;
    float* predv        = ws + o; o += (size_t)B_ * 3 * H_;
    float* tbuf         = ws + o; o += (size_t)B_ * H_;
    int*   knn          = (int*)(ws + o);  // B*KNN ints

    // 1. atom segment sums -> node_hiddens, center_pos
    seg_atoms_kernel<<<B_, H_, 0, stream>>>(h_ctx_focal, pos_ctx_focal, atoms_batch,
                                            node_hiddens, center_pos);
    // 2. prototype stats
    proto_stats_kernel<<<P_, 256, 0, stream>>>(prototypes, pcount, pnorm, inv_sp);
    vocab_coeff_kernel<<<(V_ + 255) / 256, 256, 0, stream>>>(pcount, coeff);
    // 3. masked residue segment sum + norms (dominant 256MB stream)
    seg_residue_kernel<<<B_, H_, 0, stream>>>(h_residue, residue_pos, aa_batch,
                                              center_pos, residue_emb, rnorm);
    // 4. cosine + top-4
    cos_topk_kernel<<<B_, P_, 0, stream>>>(residue_emb, prototypes, rnorm, pnorm, knn);
    // 5. prototype aggregation (A @ node_feat, prototype rows only)
    yproto_kernel<<<P_, H_, 0, stream>>>(residue_emb, knn, pcount, coeff, inv_sp, embw, Yproto);
    // 6. h1p = relu(Yproto @ gnn_w1)   [128,256]x[256,256]  (WMMA, 16x64/wave)
    {
        int waves = (P_ / 16) * (H_ / 64);        // 32
        gemm_wmma_kn_t<H_, H_, true><<<(waves * 32 + 255) / 256, 256, 0, stream>>>(
            Yproto, gnn_w1, nullptr, h1p, P_);
    }
    // 7. hp2 = h1p @ gnn_w2                              (WMMA)
    {
        int waves = (P_ / 16) * (H_ / 64);
        gemm_wmma_kn_t<H_, H_, false><<<(waves * 32 + 255) / 256, 256, 0, stream>>>(
            h1p, gnn_w2, nullptr, hp2, P_);
    }
    // 8. pred_vecs assembly
    predvec_kernel<<<B_, H_, 0, stream>>>(node_hiddens, embw, current_wid, hp2, knn, predv);
    // 9. t = pred_vecs @ motif_w + motif_b  [4096,768]x[768,256]  (WMMA)
    {
        int waves = (B_ / 16) * (H_ / 64);        // 1024
        gemm_wmma_kn_t<3 * H_, H_, false><<<(waves * 32 + 255) / 256, 256, 0, stream>>>(
            predv, motif_w, motif_b, tbuf, B_);
    }
    // 10. scores = t @ embedding_weight^T  [4096,2001]  (WMMA, guarded tail)
    {
        int Nreal   = V_ + 1;                     // 2001
        int groupsN = (Nreal + 63) / 64;          // 32
        int waves   = (B_ / 16) * groupsN;        // 8192
        gemm_wmma_nt_t<H_><<<(waves * 32 + 255) / 256, 256, 0, stream>>>(
            tbuf, embw, out, B_, Nreal, groupsN);
    }
    (void)in_sizes; (void)n_in; (void)out_size; (void)ws_size;
}